// VolumeRenderer_16063177687332
// MI455X (gfx1250) — compile-verified
//
#include <hip/hip_runtime.h>

// Volume rendering compositing for MI455X (gfx1250, wave32).
// One wave per ray (P=128 -> 4 samples/lane, float4-coalesced loads),
// multiplicative shuffle-scan for transmittance, and a
// V_WMMA_F32_16X16X4_F32 cross-lane reduction (B = ones) for the final
// 16-ray x 4-channel weighted sums, keeping full fp32 precision.

constexpr int NP = 128;          // samples per ray (reference P)
constexpr float FAR_DELTA = 1e10f;

typedef __attribute__((ext_vector_type(2))) float v2f;
typedef __attribute__((ext_vector_type(8))) float v8f;

__global__ __launch_bounds__(512)
void volrend_kernel(const float* __restrict__ density,
                    const float* __restrict__ depth,
                    const float* __restrict__ feature,
                    float* __restrict__ out, int N)
{
    // per-(wave,lane) partials: {feat.x, feat.y, feat.z, depth}; padded to
    // 33 float4 per row so phase-2 gathers stride banks by 4 (conflict-light)
    __shared__ float4 part[16][33];

    const int lane    = threadIdx.x & 31;
    const int wave    = threadIdx.x >> 5;
    const int rayBase = blockIdx.x * 16;
    const int ray     = rayBase + wave;

    float fx = 0.f, fy = 0.f, fz = 0.f, dd = 0.f;

    if (ray < N) {
        // lane owns samples p = 4*lane .. 4*lane+3 (contiguous -> b128 loads)
        const float4 sg = *(const float4*)(density + ray * NP + lane * 4);
        const float4 dp = *(const float4*)(depth   + ray * NP + lane * 4);

        // delta for last element of chunk needs next lane's first depth
        float nx = __shfl_down(dp.x, 1, 32);
        float d0 = dp.y - dp.x;
        float d1 = dp.z - dp.y;
        float d2 = dp.w - dp.z;
        float d3 = (lane == 31) ? FAR_DELTA : (nx - dp.w);

        float s0 = __expf(-sg.x * d0);
        float s1 = __expf(-sg.y * d1);
        float s2 = __expf(-sg.z * d2);
        float s3 = __expf(-sg.w * d3);

        // exclusive multiplicative scan over lanes of per-lane product
        float t = s0 * s1 * s2 * s3;
        #pragma unroll
        for (int off = 1; off < 32; off <<= 1) {
            float v = __shfl_up(t, off, 32);
            if (lane >= off) t *= v;
        }
        float excl = __shfl_up(t, 1, 32);
        if (lane == 0) excl = 1.0f;

        float T0 = excl;
        float T1 = T0 * s0;
        float T2 = T1 * s1;
        float T3 = T2 * s2;
        float w0 = T0 * (1.0f - s0);
        float w1 = T1 * (1.0f - s1);
        float w2 = T2 * (1.0f - s2);
        float w3 = T3 * (1.0f - s3);

        // feature[ray][p][0..2]: 12 floats per lane = 3 coalesced float4s
        const float* fp = feature + (size_t)ray * (NP * 3) + lane * 12;
        const float4 fa = *(const float4*)(fp + 0);   // p0.xyz, p1.x
        const float4 fb = *(const float4*)(fp + 4);   // p1.yz,  p2.xy
        const float4 fc = *(const float4*)(fp + 8);   // p2.z,   p3.xyz

        fx = w0 * fa.x + w1 * fa.w + w2 * fb.z + w3 * fc.y;
        fy = w0 * fa.y + w1 * fb.x + w2 * fb.w + w3 * fc.z;
        fz = w0 * fa.z + w1 * fb.y + w2 * fc.x + w3 * fc.w;
        dd = w0 * dp.x + w1 * dp.y + w2 * dp.z + w3 * dp.w;
    }

    part[wave][lane] = make_float4(fx, fy, fz, dd);
    __syncthreads();

    // ---- Phase 2 (wave 0 only, EXEC all ones): WMMA row-sum reduction ----
    // A[m][k] = partial of channel c, ray m, source-lane k (16x32, f32)
    // B = ones => D[m][n] = rowsum(m) for every n; chain C over 8 K-steps.
    if (wave == 0) {
        const float* Pf = (const float*)part;        // [row][k(pad33)][c]
        const v2f bones = {1.0f, 1.0f};
        const int row   = lane & 15;
        const int khalf = (lane >> 4) << 1;          // lanes 16-31 hold K+2,K+3

        #pragma unroll
        for (int c = 0; c < 4; ++c) {
            v8f acc = {0.f, 0.f, 0.f, 0.f, 0.f, 0.f, 0.f, 0.f};
            #pragma unroll
            for (int kb = 0; kb < 8; ++kb) {
                const int k0 = kb * 4 + khalf;
                v2f a;
                a[0] = Pf[(row * 33 + k0)     * 4 + c];
                a[1] = Pf[(row * 33 + k0 + 1) * 4 + c];
                acc = __builtin_amdgcn_wmma_f32_16x16x4_f32(
                    false, a, false, bones, (short)0, acc, false, false);
            }
            // D layout: VGPR j = row j (lanes 0-15), row j+8 (lanes 16-31)
            #pragma unroll
            for (int j = 0; j < 8; ++j) {
                if (lane == j) {
                    int r = rayBase + j;
                    if (r < N) {
                        if (c < 3) out[r * 3 + c]   = acc[j];
                        else       out[3 * N + r]   = acc[j];
                    }
                }
                if (lane == 16 + j) {
                    int r = rayBase + 8 + j;
                    if (r < N) {
                        if (c < 3) out[r * 3 + c]   = acc[j];
                        else       out[3 * N + r]   = acc[j];
                    }
                }
            }
        }
    }
}

extern "C" void kernel_launch(void* const* d_in, const int* in_sizes, int n_in,
                              void* d_out, int out_size, void* d_ws, size_t ws_size,
                              hipStream_t stream) {
    const float* density = (const float*)d_in[0];
    const float* depth   = (const float*)d_in[1];
    const float* feature = (const float*)d_in[2];
    float* out = (float*)d_out;

    const int N = in_sizes[0] / NP;              // rays
    const int blocks = (N + 15) / 16;            // 16 rays (16 waves) per block

    hipLaunchKernelGGL(volrend_kernel, dim3(blocks), dim3(512), 0, stream,
                       density, depth, feature, out, N);
}